// TriangleAttention_68178310857258
// MI455X (gfx1250) — compile-verified
//
#include <hip/hip_runtime.h>
#include <math.h>

typedef __attribute__((ext_vector_type(2))) float v2f;
typedef __attribute__((ext_vector_type(8))) float v8f;

#define NSEQ   128
#define CDIM   128
#define HHEADS 8
#define DHEAD  16
#define QSCALE 0.25f   // D^-0.5, D=16

// ---- fp32 WMMA 16x16x4 helper (D = A(16x4) * B(4x16) + C) -----------------
// A layout: lane m=L%16 holds row m; half=L/16 picks K pair {2*half, 2*half+1}
// B layout: lane n=L%16 holds col n; half picks K pair (mirrors C/D layout)
// C/D layout: VGPR v, lanes0-15 -> M=v, lanes16-31 -> M=v+8, N = L%16
__device__ __forceinline__ v8f wmma4(v2f a, v2f b, v8f c) {
  return __builtin_amdgcn_wmma_f32_16x16x4_f32(false, a, false, b,
                                               (short)0, c, false, false);
}

// ---------------------------------------------------------------------------
// LayerNorm over last dim (128). One wave32 per row, 4 elems/lane.
// ---------------------------------------------------------------------------
__global__ __launch_bounds__(256) void ln_kernel(
    const float* __restrict__ e, const float* __restrict__ w,
    const float* __restrict__ b, float* __restrict__ out, int rows) {
  int wave = (int)((blockIdx.x * blockDim.x + threadIdx.x) >> 5);
  int lane = threadIdx.x & 31;
  if (wave >= rows) return;
  const float4 v  = ((const float4*)(e + (size_t)wave * CDIM))[lane];
  float s  = v.x + v.y + v.z + v.w;
  float ss = v.x * v.x + v.y * v.y + v.z * v.z + v.w * v.w;
  for (int o = 16; o > 0; o >>= 1) {
    s  += __shfl_xor(s,  o, 32);
    ss += __shfl_xor(ss, o, 32);
  }
  float mean = s * (1.0f / CDIM);
  float var  = ss * (1.0f / CDIM) - mean * mean;
  float r    = rsqrtf(var + 1e-5f);
  float4 wv = ((const float4*)w)[lane];
  float4 bv = ((const float4*)b)[lane];
  float4 o4;
  o4.x = (v.x - mean) * r * wv.x + bv.x;
  o4.y = (v.y - mean) * r * wv.y + bv.y;
  o4.z = (v.z - mean) * r * wv.z + bv.z;
  o4.w = (v.w - mean) * r * wv.w + bv.w;
  ((float4*)(out + (size_t)wave * CDIM))[lane] = o4;
}

// ---------------------------------------------------------------------------
// Generic C = A[M,K] * B[K,N] + bias (row-major).
// Each wave: one 16-row M-tile x NT 16-col N-tiles, two K accumulator chains.
// M multiple of 16, N multiple of 16*NT, K multiple of 8. 8 waves / block.
// All addressing uses 32-bit element offsets (buffers < 2^31 elements).
// ---------------------------------------------------------------------------
template <int NT>
__global__ __launch_bounds__(256) void gemm_wmma(
    const float* __restrict__ A, const float* __restrict__ B,
    const float* __restrict__ bias, float* __restrict__ C,
    int M, int N, int K) {
  int wave = threadIdx.x >> 5;
  int lane = threadIdx.x & 31;
  int half = lane >> 4, l16 = lane & 15;
  int groups_n = N / (16 * NT);
  int tile = blockIdx.x * 8 + wave;
  int tm = tile / groups_n;
  int tn = (tile - tm * groups_n) * NT;
  if (tm * 16 >= M) return;

  const int aoff = (tm * 16 + l16) * K;   // A row base (element offset)
  const int h2   = 2 * half;
  v8f acc[NT][2];
  for (int u = 0; u < NT; ++u) {
    acc[u][0] = (v8f){};
    acc[u][1] = (v8f){};
  }
  for (int k = 0; k < K; k += 8) {
    v2f a0, a1;
    a0.x = A[aoff + k + h2];
    a0.y = A[aoff + k + h2 + 1];
    a1.x = A[aoff + k + 4 + h2];
    a1.y = A[aoff + k + 4 + h2 + 1];
    if (k + 8 < K)  // hint next B k-block into cache (global_prefetch_b8)
      __builtin_prefetch(B + (k + 8) * N + tn * 16 + l16, 0, 0);
    for (int u = 0; u < NT; ++u) {
      const int boff = (tn + u) * 16 + l16;
      v2f b0, b1;
      b0.x = B[(k + h2) * N + boff];
      b0.y = B[(k + h2 + 1) * N + boff];
      b1.x = B[(k + 4 + h2) * N + boff];
      b1.y = B[(k + 4 + h2 + 1) * N + boff];
      acc[u][0] = wmma4(a0, b0, acc[u][0]);
      acc[u][1] = wmma4(a1, b1, acc[u][1]);
    }
  }
  for (int u = 0; u < NT; ++u) {
    const int bcol = (tn + u) * 16 + l16;
    float bv = bias ? bias[bcol] : 0.0f;
    for (int v = 0; v < 8; ++v) {
      int row = tm * 16 + v + 8 * half;
      C[row * N + bcol] = acc[u][0][v] + acc[u][1][v] + bv;
    }
  }
}

// ---------------------------------------------------------------------------
// Fused triangle attention for one orientation.
//   grid.x = 128*8: j = blockIdx.x>>3, i-tile = blockIdx.x&7.
//   8 waves/block, wave = head h.
// Per wave (16 query rows):
//   logits[16,128] = (Q*SCALE) @ K^T   (8 independent WMMA chains)
//   bias pass: acc += E + mask; sigmoid(G + mask) stashed in the LDS strip
//   softmax over k; probs = acc * inv * gate (gate re-read from LDS),
//   final probs overwrite the LDS strip; Va[16,16] = probs @ V (K=128, 2 chains)
// out_mode==0 ("in"):  K/V element stride over k = 384      (rows j*N+k),
//                      bias/gate/mask at (i*N+k)
// out_mode==1 ("out"): K/V element stride over k = 128*384  (rows k*N+j),
//                      bias/gate/mask at (k*N+i)
// All addressing uses 32-bit element offsets from the base pointers.
// ---------------------------------------------------------------------------
__global__ __launch_bounds__(256) void attn_kernel(
    const float* __restrict__ qkv,   // [N*N, 384]  (Q|K|V, col = d*8+h)
    const float* __restrict__ eg,    // [N*N, 16]   (E cols 0..7, G cols 8..15)
    const float* __restrict__ mask,  // [N*N]
    float* __restrict__ Va,          // [N*N, 256]
    int out_mode) {
  __shared__ float lds[HHEADS * 16 * NSEQ];   // 8 waves x 16x128 f32 = 64 KB
  const int j    = blockIdx.x >> 3;
  const int i0   = (blockIdx.x & 7) * 16;
  const int h    = threadIdx.x >> 5;
  const int lane = threadIdx.x & 31;
  const int half = lane >> 4, l16 = lane & 15;
  float* strip = lds + h * 16 * NSEQ;   // per-wave 16x128 scratch

  // K/V addressing: element(k) = qkv[off + k*kstride]
  const int kstride = out_mode ? NSEQ * 384 : 384;
  const int jbase   = out_mode ? j * 384 : j * NSEQ * 384;
  const int koff    = jbase + 128 + h;           // + kg*kstride + d*8
  const int voff    = jbase + 256 + l16 * 8 + h; // + k*kstride

  // ---- hoist the 4 Q fragments (shared by all 8 k-tiles) ----
  const int qoff = ((i0 + l16) * NSEQ + j) * 384 + h;
  v2f q[4];
  for (int s = 0; s < 4; ++s) {
    int d0 = 4 * s + 2 * half;
    q[s].x = qkv[qoff + d0 * 8] * QSCALE;
    q[s].y = qkv[qoff + (d0 + 1) * 8] * QSCALE;
  }

  // ---- logits: 8 accumulator tiles covering k = 0..127 ----
  v8f acc[8];
  int bkoff[8];
  for (int t = 0; t < 8; ++t) {
    acc[t] = (v8f){};
    bkoff[t] = koff + (t * 16 + l16) * kstride;
  }
  for (int s = 0; s < 4; ++s) {
    int d0 = 4 * s + 2 * half;
    for (int t = 0; t < 8; ++t) {
      v2f b;
      b.x = qkv[bkoff[t] + d0 * 8];
      b.y = qkv[bkoff[t] + (d0 + 1) * 8];
      acc[t] = wmma4(q[s], b, acc[t]);
    }
  }

  // ---- bias pass: acc += E + mask; gate = sigmoid(G + mask) -> LDS strip ----
  for (int t = 0; t < 8; ++t) {
    const int kg = t * 16 + l16;
    for (int v = 0; v < 8; ++v) {
      const int ig = i0 + v + 8 * half;
      const int idx2 = out_mode ? (kg * NSEQ + ig) : (ig * NSEQ + kg);
      const int eb = idx2 * 16 + h;
      float mk = mask[idx2];
      acc[t][v] += eg[eb] + mk;
      float g = 1.0f / (1.0f + expf(-(eg[eb + 8] + mk)));
      strip[(v + 8 * half) * NSEQ + kg] = g;
    }
  }

  // ---- row softmax over k: reduce across 8 tiles + 16 lanes per half ----
  float inv[8];
  for (int v = 0; v < 8; ++v) {
    float m = acc[0][v];
    for (int t = 1; t < 8; ++t) m = fmaxf(m, acc[t][v]);
    for (int o = 8; o > 0; o >>= 1) m = fmaxf(m, __shfl_xor(m, o, 16));
    float s = 0.0f;
    for (int t = 0; t < 8; ++t) {
      float e = expf(acc[t][v] - m);
      acc[t][v] = e;
      s += e;
    }
    for (int o = 8; o > 0; o >>= 1) s += __shfl_xor(s, o, 16);
    inv[v] = 1.0f / s;
  }
  asm volatile("s_wait_dscnt 0" ::: "memory");  // gate stores done

  // ---- apply gate (from LDS) + overwrite strip with final probs ----
  for (int t = 0; t < 8; ++t) {
    const int kg = t * 16 + l16;
    for (int v = 0; v < 8; ++v) {
      const int sa = (v + 8 * half) * NSEQ + kg;
      strip[sa] = acc[t][v] * inv[v] * strip[sa];
    }
  }
  asm volatile("s_wait_dscnt 0" ::: "memory");  // prob stores done

  // ---- Va tile = probs[16,128] @ V[128,16] (WMMA, K=128, 2 chains) ----
  v8f ov0 = {}, ov1 = {};
  for (int kk = 0; kk < NSEQ; kk += 8) {
    const int k0 = kk + 2 * half;
    const int k1 = kk + 4 + 2 * half;
    v2f a0, a1, b0, b1;
    a0.x = strip[l16 * NSEQ + k0];
    a0.y = strip[l16 * NSEQ + k0 + 1];
    a1.x = strip[l16 * NSEQ + k1];
    a1.y = strip[l16 * NSEQ + k1 + 1];
    b0.x = qkv[voff + k0 * kstride];
    b0.y = qkv[voff + (k0 + 1) * kstride];
    b1.x = qkv[voff + k1 * kstride];
    b1.y = qkv[voff + (k1 + 1) * kstride];
    ov0 = wmma4(a0, b0, ov0);
    ov1 = wmma4(a1, b1, ov1);
  }
  const int obase = out_mode * 128 + l16 * 8 + h;
  for (int v = 0; v < 8; ++v) {
    const int ig = i0 + v + 8 * half;
    Va[(ig * NSEQ + j) * 256 + obase] = ov0[v] + ov1[v];
  }
}

// ---------------------------------------------------------------------------
extern "C" void kernel_launch(void* const* d_in, const int* in_sizes, int n_in,
                              void* d_out, int out_size, void* d_ws,
                              size_t ws_size, hipStream_t stream) {
  (void)in_sizes; (void)n_in; (void)out_size; (void)ws_size;
  const float* e        = (const float*)d_in[0];
  const float* mask     = (const float*)d_in[1];
  const float* ln_w     = (const float*)d_in[2];
  const float* ln_b     = (const float*)d_in[3];
  const float* w_qkv_in = (const float*)d_in[4];
  const float* b_qkv_in = (const float*)d_in[5];
  const float* w_eg_in  = (const float*)d_in[6];
  const float* b_eg_in  = (const float*)d_in[7];
  const float* w_qkv_o  = (const float*)d_in[8];
  const float* b_qkv_o  = (const float*)d_in[9];
  const float* w_eg_o   = (const float*)d_in[10];
  const float* b_eg_o   = (const float*)d_in[11];
  const float* w_o      = (const float*)d_in[12];
  const float* b_o      = (const float*)d_in[13];
  float* out = (float*)d_out;

  const int ROWS = NSEQ * NSEQ;                   // 16384
  float* e_ln = (float*)d_ws;                     //  8 MB
  float* qkv  = e_ln + (size_t)ROWS * CDIM;       // 24 MB (reused in/out)
  float* eg   = qkv  + (size_t)ROWS * 3 * CDIM;   //  1 MB (reused in/out)
  float* Va   = eg   + (size_t)ROWS * 16;         // 16 MB [ROWS, 256]

  // 1. LayerNorm
  ln_kernel<<<ROWS / 8, 256, 0, stream>>>(e, ln_w, ln_b, e_ln, ROWS);

  // 2. "in" projections + fused attention -> Va[:, :128]
  gemm_wmma<2><<<(ROWS / 16) * (384 / 32) / 8, 256, 0, stream>>>(
      e_ln, w_qkv_in, b_qkv_in, qkv, ROWS, 384, CDIM);
  gemm_wmma<1><<<(ROWS / 16) * (16 / 16) / 8, 256, 0, stream>>>(
      e_ln, w_eg_in, b_eg_in, eg, ROWS, 16, CDIM);
  attn_kernel<<<NSEQ * 8, 256, 0, stream>>>(qkv, eg, mask, Va, 0);

  // 3. "out" projections + fused attention -> Va[:, 128:]
  gemm_wmma<2><<<(ROWS / 16) * (384 / 32) / 8, 256, 0, stream>>>(
      e_ln, w_qkv_o, b_qkv_o, qkv, ROWS, 384, CDIM);
  gemm_wmma<1><<<(ROWS / 16) * (16 / 16) / 8, 256, 0, stream>>>(
      e_ln, w_eg_o, b_eg_o, eg, ROWS, 16, CDIM);
  attn_kernel<<<NSEQ * 8, 256, 0, stream>>>(qkv, eg, mask, Va, 1);

  // 4. Output projection: out = Va[ROWS,256] @ w_o[256,128] + b_o
  gemm_wmma<2><<<(ROWS / 16) * (128 / 32) / 8, 256, 0, stream>>>(
      Va, w_o, b_o, out, ROWS, CDIM, 2 * CDIM);
}